// QuadrotorControllerModule_5068061409551
// MI455X (gfx1250) — compile-verified
//
#include <hip/hip_runtime.h>
#include <math.h>

// ---- problem constants (match reference) ----
#define T_STEPS 400
#define BATCH   4096
#define BLOCK   128
#define SDIM    13

#define GCONST 9.81f
#define MASSC  0.034f
#define ARMC   (0.707106781f * 0.046f)
#define T2TC   0.006f
#define DTC    0.01f
#define EPSC   1e-8f

#if __has_builtin(__builtin_amdgcn_global_load_async_to_lds_b32) && \
    __has_builtin(__builtin_amdgcn_s_wait_asynccnt)
#define USE_ASYNC_LDS 1
#else
#define USE_ASYNC_LDS 0
#endif

#if USE_ASYNC_LDS
typedef __attribute__((address_space(1))) int* gas1_t;
typedef __attribute__((address_space(3))) int* las3_t;
#endif

__global__ __launch_bounds__(BLOCK) void quad_ctrl_kernel(
    const float* __restrict__ sp,
    const float* __restrict__ kp_p, const float* __restrict__ kv_p,
    const float* __restrict__ kw_p, const float* __restrict__ kr_p,
    float* __restrict__ outY, float* __restrict__ outRd, float* __restrict__ outW)
{
    const int tid = threadIdx.x;
    const int b   = blockIdx.x * BLOCK + tid;

    const float kp = *kp_p, kv = *kv_p, kw = *kw_p, kr = *kr_p;

    const float J0 = 1.657171e-05f, J1 = 1.6655602e-05f, J2 = 2.9261652e-05f;
    const float C1 = 1.0f / (4.0f * ARMC);   // analytic inv(B0) coefficients
    const float C2 = 1.0f / (4.0f * T2TC);

    // state registers
    float p0=0.f,p1=0.f,p2_=0.f, v0=0.f,v1=0.f,v2_=0.f;
    float qw=1.f,qx=0.f,qy=0.f,qz=0.f, w0=0.f,w1=0.f,w2_=0.f;

#if USE_ASYNC_LDS
    __shared__ float smem[2][BLOCK/32][32 * SDIM];   // per-wave double-buffered tiles
    const int wid  = tid >> 5;
    const int lane = tid & 31;
    {   // prologue: async-DMA step 0 setpoints for this wave into buffer 0
        const float* g = sp + ((size_t)blockIdx.x * BLOCK) * SDIM + (size_t)wid * (32 * SDIM);
        #pragma unroll
        for (int j = 0; j < SDIM; ++j) {
            __builtin_amdgcn_global_load_async_to_lds_b32(
                (gas1_t)(uintptr_t)(g + j * 32 + lane),
                (las3_t)(&smem[0][wid][j * 32 + lane]), 0, 0);
        }
    }
#endif

    for (int t = 0; t < T_STEPS; ++t) {
        float s[SDIM];
#if USE_ASYNC_LDS
        __builtin_amdgcn_s_wait_asynccnt(0);
        asm volatile("" ::: "memory");
        {
            const float* l = &smem[t & 1][wid][lane * SDIM];
            #pragma unroll
            for (int i = 0; i < SDIM; ++i) s[i] = l[i];
        }
        if (t + 1 < T_STEPS) {   // overlap next step's DMA with this step's math
            const float* g = sp + ((size_t)(t + 1) * BATCH + (size_t)blockIdx.x * BLOCK) * SDIM
                               + (size_t)wid * (32 * SDIM);
            float* lb = &smem[(t + 1) & 1][wid][0];
            #pragma unroll
            for (int j = 0; j < SDIM; ++j) {
                __builtin_amdgcn_global_load_async_to_lds_b32(
                    (gas1_t)(uintptr_t)(g + j * 32 + lane),
                    (las3_t)(lb + j * 32 + lane), 0, 0);
            }
        }
#else
        {
            const float* g = sp + ((size_t)t * BATCH + b) * SDIM;
            if (t + 1 < T_STEPS)
                __builtin_prefetch(sp + ((size_t)(t + 1) * BATCH + b) * SDIM, 0, 0);
            #pragma unroll
            for (int i = 0; i < SDIM; ++i) s[i] = g[i];
        }
#endif
        const float pd0=s[0], pd1=s[1], pd2=s[2];
        const float vd0=s[3], vd1=s[4], vd2=s[5];
        const float ad0=s[6], ad1=s[7], ad2=s[8];
        const float jd0=s[9], jd1=s[10], jd2=s[11];
        const float yawd=s[12];

        if (t == 0) { p0=pd0; p1=pd1; p2_=pd2; v0=vd0; v1=vd1; v2_=vd2; }

        // ---- emit ys[t] = state BEFORE the step (scan emits the carry-in) ----
        {
            float* y = outY + ((size_t)t * BATCH + b) * SDIM;
            __builtin_nontemporal_store(p0,  y + 0);
            __builtin_nontemporal_store(p1,  y + 1);
            __builtin_nontemporal_store(p2_, y + 2);
            __builtin_nontemporal_store(v0,  y + 3);
            __builtin_nontemporal_store(v1,  y + 4);
            __builtin_nontemporal_store(v2_, y + 5);
            __builtin_nontemporal_store(qw,  y + 6);
            __builtin_nontemporal_store(qx,  y + 7);
            __builtin_nontemporal_store(qy,  y + 8);
            __builtin_nontemporal_store(qz,  y + 9);
            __builtin_nontemporal_store(w0,  y + 10);
            __builtin_nontemporal_store(w1,  y + 11);
            __builtin_nontemporal_store(w2_, y + 12);
        }

        // ---- rotation matrix from quaternion ----
        const float r00 = 1.f - 2.f*(qy*qy + qz*qz);
        const float r01 = 2.f*(qx*qy - qz*qw);
        const float r02 = 2.f*(qx*qz + qy*qw);
        const float r10 = 2.f*(qx*qy + qz*qw);
        const float r11 = 1.f - 2.f*(qx*qx + qz*qz);
        const float r12 = 2.f*(qy*qz - qx*qw);
        const float r20 = 2.f*(qx*qz - qy*qw);
        const float r21 = 2.f*(qy*qz + qx*qw);
        const float r22 = 1.f - 2.f*(qx*qx + qy*qy);
        // z_b = (r02, r12, r22)

        // ---- desired force & attitude ----
        const float Fd0 = MASSC*ad0            + kp*(pd0 - p0)  + kv*(vd0 - v0);
        const float Fd1 = MASSC*ad1            + kp*(pd1 - p1)  + kv*(vd1 - v1);
        const float Fd2 = MASSC*(ad2 + GCONST) + kp*(pd2 - p2_) + kv*(vd2 - v2_);
        const float thrust = Fd0*r02 + Fd1*r12 + Fd2*r22;

        const float nF = sqrtf(Fd0*Fd0 + Fd1*Fd1 + Fd2*Fd2);
        const float iF = 1.0f / (nF + EPSC);
        const float zd0 = Fd0*iF, zd1 = Fd1*iF, zd2 = Fd2*iF;

        float sy, cy;
        __sincosf(yawd, &sy, &cy);
        // cross(zdes, xc), xc = (cy, sy, 0)
        const float cx0 = -zd2*sy;
        const float cx1 =  zd2*cy;
        const float cx2 =  zd0*sy - zd1*cy;
        const float nY  = sqrtf(cx0*cx0 + cx1*cx1 + cx2*cx2);
        const float iY  = 1.0f / (nY + EPSC);
        const float yd0 = cx0*iY, yd1 = cx1*iY, yd2 = cx2*iY;
        // xdes = cross(ydes, zdes)
        const float xd0 = yd1*zd2 - yd2*zd1;
        const float xd1 = yd2*zd0 - yd0*zd2;
        const float xd2 = yd0*zd1 - yd1*zd0;

        // ---- emit Rds[t] (rows of Rd = [xdes ydes zdes] as columns) ----
        {
            float* rd = outRd + ((size_t)t * BATCH + b) * 9;
            __builtin_nontemporal_store(xd0, rd + 0);
            __builtin_nontemporal_store(yd0, rd + 1);
            __builtin_nontemporal_store(zd0, rd + 2);
            __builtin_nontemporal_store(xd1, rd + 3);
            __builtin_nontemporal_store(yd1, rd + 4);
            __builtin_nontemporal_store(zd1, rd + 5);
            __builtin_nontemporal_store(xd2, rd + 6);
            __builtin_nontemporal_store(yd2, rd + 7);
            __builtin_nontemporal_store(zd2, rd + 8);
        }

        // ---- attitude error eR = 0.5*vee(M - M^T), M = Rd^T R ----
        const float M21 = zd0*r01 + zd1*r11 + zd2*r21;
        const float M12 = yd0*r02 + yd1*r12 + yd2*r22;
        const float M02 = xd0*r02 + xd1*r12 + xd2*r22;
        const float M20 = zd0*r00 + zd1*r10 + zd2*r20;
        const float M10 = yd0*r00 + yd1*r10 + yd2*r20;
        const float M01 = xd0*r01 + xd1*r11 + xd2*r21;
        const float eR0 = 0.5f*(M21 - M12);
        const float eR1 = 0.5f*(M02 - M20);
        const float eR2 = 0.5f*(M10 - M01);

        // ---- desired body rate ----
        const float zjd = zd0*jd0 + zd1*jd1 + zd2*jd2;
        const float hs  = MASSC / (thrust + EPSC);
        const float hw0 = hs*(jd0 - zjd*zd0);
        const float hw1 = hs*(jd1 - zjd*zd1);
        const float hw2 = hs*(jd2 - zjd*zd2);
        const float dW0 = -(hw0*yd0 + hw1*yd1 + hw2*yd2);
        const float dW1 =  (hw0*xd0 + hw1*xd1 + hw2*xd2);

        {
            float* dw = outW + ((size_t)t * BATCH + b) * 3;
            __builtin_nontemporal_store(dW0, dw + 0);
            __builtin_nontemporal_store(dW1, dw + 1);
            __builtin_nontemporal_store(0.f, dw + 2);
        }

        // eW = w - R^T (Rd @ desW)
        const float u0_ = xd0*dW0 + yd0*dW1;
        const float u1_ = xd1*dW0 + yd1*dW1;
        const float u2_ = xd2*dW0 + yd2*dW1;
        const float e0  = r00*u0_ + r10*u1_ + r20*u2_;
        const float e1  = r01*u0_ + r11*u1_ + r21*u2_;
        const float e2  = r02*u0_ + r12*u1_ + r22*u2_;
        const float eW0 = w0 - e0, eW1 = w1 - e1, eW2 = w2_ - e2;

        const float Jw0 = J0*w0, Jw1 = J1*w1, Jw2 = J2*w2_;
        const float cwJ0 = w1*Jw2 - w2_*Jw1;
        const float cwJ1 = w2_*Jw0 - w0*Jw2;
        const float cwJ2 = w0*Jw1 - w1*Jw0;
        const float tq0 = -kr*eR0 - kw*eW0 + cwJ0;
        const float tq1 = -kr*eR1 - kw*eW1 + cwJ1;
        const float tq2 = -kr*eR2 - kw*eW2 + cwJ2;

        // ---- mixer: force = B0inv @ wrench; eta = B0 @ force ----
        const float qT = 0.25f*thrust;
        const float a_ = C1*tq0, b_ = C1*tq1, c_ = C2*tq2;
        const float f1_ = qT - a_ - b_ - c_;
        const float f2_ = qT - a_ + b_ + c_;
        const float f3_ = qT + a_ + b_ - c_;
        const float f4_ = qT + a_ - b_ + c_;
        const float eta0 = f1_ + f2_ + f3_ + f4_;
        const float eta1 = ARMC*(-f1_ - f2_ + f3_ + f4_);
        const float eta2 = ARMC*(-f1_ + f2_ + f3_ - f4_);
        const float eta3 = T2TC*(-f1_ + f2_ - f3_ + f4_);

        // ---- integrate ----
        const float fm   = eta0 / MASSC;
        const float acc0 = fm*r02;
        const float acc1 = fm*r12;
        const float acc2 = fm*r22 - GCONST;

        const float np0 = p0 + DTC*v0;
        const float np1 = p1 + DTC*v1;
        const float np2 = p2_ + DTC*v2_;
        const float nv0 = v0 + DTC*acc0;
        const float nv1 = v1 + DTC*acc1;
        const float nv2 = v2_ + DTC*acc2;

        const float dqw = -0.5f*(qx*w0 + qy*w1 + qz*w2_);
        const float dqx =  0.5f*(qw*w0 + (qy*w2_ - qz*w1));
        const float dqy =  0.5f*(qw*w1 + (qz*w0 - qx*w2_));
        const float dqz =  0.5f*(qw*w2_ + (qx*w1 - qy*w0));
        const float tqw = qw + DTC*dqw;
        const float tqx = qx + DTC*dqx;
        const float tqy = qy + DTC*dqy;
        const float tqz = qz + DTC*dqz;
        const float qn  = sqrtf(tqw*tqw + tqx*tqx + tqy*tqy + tqz*tqz);
        const float iq  = 1.0f / (qn + EPSC);

        const float nw0 = w0  + DTC*((eta1 - cwJ0) / J0);
        const float nw1 = w1  + DTC*((eta2 - cwJ1) / J1);
        const float nw2 = w2_ + DTC*((eta3 - cwJ2) / J2);

        p0 = np0; p1 = np1; p2_ = np2;
        v0 = nv0; v1 = nv1; v2_ = nv2;
        qw = tqw*iq; qx = tqx*iq; qy = tqy*iq; qz = tqz*iq;
        w0 = nw0; w1 = nw1; w2_ = nw2;
    }
}

extern "C" void kernel_launch(void* const* d_in, const int* in_sizes, int n_in,
                              void* d_out, int out_size, void* d_ws, size_t ws_size,
                              hipStream_t stream) {
    (void)in_sizes; (void)n_in; (void)out_size; (void)d_ws; (void)ws_size;
    const float* sp = (const float*)d_in[0];
    const float* kp = (const float*)d_in[1];
    const float* kv = (const float*)d_in[2];
    const float* kw = (const float*)d_in[3];
    const float* kr = (const float*)d_in[4];
    float* outY  = (float*)d_out;
    float* outRd = outY  + (size_t)T_STEPS * BATCH * SDIM;
    float* outW  = outRd + (size_t)T_STEPS * BATCH * 9;
    dim3 grid(BATCH / BLOCK), block(BLOCK);
    hipLaunchKernelGGL(quad_ctrl_kernel, grid, block, 0, stream,
                       sp, kp, kv, kw, kr, outY, outRd, outW);
}